// SoftArgmax_51221779972400
// MI455X (gfx1250) — compile-verified
//
#include <hip/hip_runtime.h>
#include <hip/hip_bf16.h>

// Problem constants (from reference): heatmap (32, 64, 256, 256) f32
#define NCH   2048      // 32*64 channels
#define HW    65536     // 256*256
#define WDIM  256
#define HDIM  256

typedef __attribute__((ext_vector_type(2))) float v2f;
typedef __attribute__((ext_vector_type(8))) float v8f;

// ---------------------------------------------------------------------------
// Kernel 1: streaming argmax per channel. This is the bandwidth-bound part:
// 512 MiB read once via b128 loads => ~22us floor at 23.3 TB/s.
// One 256-thread block per channel; each thread scans 64 float4s.
// Tie-break: lowest flat index (matches jnp.argmax first-occurrence).
// ---------------------------------------------------------------------------
__global__ __launch_bounds__(256) void softargmax_argmax_kernel(
    const float* __restrict__ H, int* __restrict__ outIdx, float* __restrict__ outVal) {
  const int ch  = blockIdx.x;
  const int tid = threadIdx.x;
  const float4* __restrict__ p = (const float4*)(H + (size_t)ch * HW);

  float best = -__builtin_inff();
  int   bidx = 0;

#pragma unroll 8
  for (int j = 0; j < 64; ++j) {
    const int v4 = tid + j * 256;           // coalesced across the block
    float4 v = p[v4];
    const int base = v4 * 4;
    if (v.x > best) { best = v.x; bidx = base + 0; }
    if (v.y > best) { best = v.y; bidx = base + 1; }
    if (v.z > best) { best = v.z; bidx = base + 2; }
    if (v.w > best) { best = v.w; bidx = base + 3; }
  }

  __shared__ float sval[256];
  __shared__ int   sidx[256];
  sval[tid] = best;
  sidx[tid] = bidx;
  __syncthreads();

#pragma unroll
  for (int s = 128; s > 0; s >>= 1) {
    if (tid < s) {
      const float v2 = sval[tid + s];
      const int   i2 = sidx[tid + s];
      if (v2 > sval[tid] || (v2 == sval[tid] && i2 < sidx[tid])) {
        sval[tid] = v2;
        sidx[tid] = i2;
      }
    }
    __syncthreads();
  }
  if (tid == 0) {
    outIdx[ch] = sidx[0];
    outVal[ch] = sval[0];
  }
}

// ---------------------------------------------------------------------------
// Kernel 2: one wave32 per channel. The <=16x16 softmax window is one WMMA
// tile. Compute D = P(16x16) x M(16x16) with M columns [1, x, x^2, 0...]
// via 4 x V_WMMA_F32_16X16X4_F32 (K accumulated over the 16 window columns).
//   D[y,0] = py[y], D[y,1] = sum_x p*x, D[y,2] = sum_x p*x^2
// Then fold rows with y-weights per lane (f32 C/D layout: lanes 0-15 hold
// rows 0-7 in v0..v7, lanes 16-31 hold rows 8-15) and 12 shuffles finish the
// scalar moments. All 32 lanes convergent at the WMMA (EXEC all 1s).
// ---------------------------------------------------------------------------
__global__ __launch_bounds__(128) void softargmax_moments_kernel(
    const float* __restrict__ H, const int* __restrict__ argIdx,
    const float* __restrict__ argVal, float* __restrict__ out) {
  const int wave = threadIdx.x >> 5;
  const int lane = threadIdx.x & 31;
  const int ch   = blockIdx.x * 4 + wave;   // 512 blocks * 4 waves = 2048

  const int   fidx = argIdx[ch];
  const float vmax = argVal[ch];
  const int x0 = fidx & (WDIM - 1);
  const int y0 = fidx >> 8;

  const int xmin = max(x0 - 8, 0);
  const int xmax = min(x0 + 8, WDIM);
  const int ymin = max(y0 - 8, 0);
  const int ymax = min(y0 + 8, HDIM);

  const float* __restrict__ base = H + (size_t)ch * HW;

  const int lo = lane & 15;   // A: row M index; B: output column N index
  const int hi = lane >> 4;   // selects K-pair within each 4-wide chunk

  const int y      = ymin + lo;            // A-matrix row -> heatmap y
  const bool yval  = (y < ymax);
  const int yclamp = min(y, HDIM - 1);
  const float* __restrict__ rowp = base + yclamp * WDIM;

  v8f acc = {};

#pragma unroll
  for (int k = 0; k < 4; ++k) {
    const int c0 = 4 * k + 2 * hi;         // window-column (K) index pair base
    const int xA = xmin + c0;
    const int xB = xA + 1;

    // A-matrix (P): masked exp((h - vmax)/T), T = 0.01
    const float h0 = rowp[min(xA, WDIM - 1)];
    const float h1 = rowp[min(xB, WDIM - 1)];
    const float a0 = (yval && xA < xmax) ? __expf(100.0f * (h0 - vmax)) : 0.0f;
    const float a1 = (yval && xB < xmax) ? __expf(100.0f * (h1 - vmax)) : 0.0f;
    v2f A = {a0, a1};

    // B-matrix (M): M[c, j] with j = lo; columns j = [1, x, x^2], rest 0.
    const float xa = (float)xA, xb = (float)xB;
    const float b0 = (lo == 0) ? 1.0f : (lo == 1) ? xa : (lo == 2) ? xa * xa : 0.0f;
    const float b1 = (lo == 0) ? 1.0f : (lo == 1) ? xb : (lo == 2) ? xb * xb : 0.0f;
    v2f Bv = {b0, b1};

    acc = __builtin_amdgcn_wmma_f32_16x16x4_f32(
        /*neg_a=*/false, A, /*neg_b=*/false, Bv,
        /*c_mod=*/(short)0, acc, /*reuse_a=*/false, /*reuse_b=*/false);
  }

  // Fold the 8 rows this lane holds, with y-weights. Rows beyond the clamped
  // window are all-zero in P, so no masking needed here.
  float s = 0.0f, sy = 0.0f, syy = 0.0f;
#pragma unroll
  for (int v = 0; v < 8; ++v) {
    const float yw = (float)(ymin + v + 8 * hi);
    const float d  = acc[v];
    s   += d;
    sy  += yw * d;
    syy += yw * yw * d;
  }

  // Column j of D lives in lanes {j, j+16}. Gather the six scalar moments.
  const float S   = __shfl(s,   0, 32) + __shfl(s,   16, 32);
  const float Sy  = __shfl(sy,  0, 32) + __shfl(sy,  16, 32);
  const float Syy = __shfl(syy, 0, 32) + __shfl(syy, 16, 32);
  const float Sx  = __shfl(s,   1, 32) + __shfl(s,   17, 32);
  const float Sxy = __shfl(sy,  1, 32) + __shfl(sy,  17, 32);
  const float Sxx = __shfl(s,   2, 32) + __shfl(s,   18, 32);

  if (lane == 0) {
    const float inv = 1.0f / S;
    const float xm  = Sx * inv;
    const float ym  = Sy * inv;
    const float vxx = Sxx * inv - xm * xm;
    const float vyy = Syy * inv - ym * ym;
    const float cxy = Sxy * inv - xm * ym;

    // Output layout: coords (2048*2) | cov (2048*4) | spread (2048*1)
    out[ch * 2 + 0] = xm * (1.0f / (WDIM - 1));
    out[ch * 2 + 1] = ym * (1.0f / (HDIM - 1));
    float* cov = out + NCH * 2;
    cov[ch * 4 + 0] = vxx;
    cov[ch * 4 + 1] = cxy;
    cov[ch * 4 + 2] = cxy;
    cov[ch * 4 + 3] = vyy;
    float* spread = out + NCH * 6;
    spread[ch] = vxx + vyy;
  }
}

extern "C" void kernel_launch(void* const* d_in, const int* in_sizes, int n_in,
                              void* d_out, int out_size, void* d_ws, size_t ws_size,
                              hipStream_t stream) {
  (void)in_sizes; (void)n_in; (void)out_size; (void)ws_size;
  const float* heatmap = (const float*)d_in[0];
  float* out = (float*)d_out;

  int*   wsIdx = (int*)d_ws;                 // 2048 ints
  float* wsVal = (float*)d_ws + NCH;         // 2048 floats (16 KB total)

  softargmax_argmax_kernel<<<NCH, 256, 0, stream>>>(heatmap, wsIdx, wsVal);
  softargmax_moments_kernel<<<NCH / 4, 128, 0, stream>>>(heatmap, wsIdx, wsVal, out);
}